// GINEInvariant_58171037057265
// MI455X (gfx1250) — compile-verified
//
#include <hip/hip_runtime.h>
#include <hip/hip_bf16.h>

typedef __attribute__((ext_vector_type(2))) float v2f;
typedef __attribute__((ext_vector_type(8))) float v8f;

static constexpr int N_ = 100000;
static constexpr int E_ = 800000;
static constexpr int H_ = 64;
static constexpr int B_ = 100;
static constexpr int L_ = 4;

#define WMMA_F32X4(a, b, c) \
  __builtin_amdgcn_wmma_f32_16x16x4_f32(false, (a), false, (b), (short)0, (c), false, false)

// ---------------------------------------------------------------------------
// Precompute invariant complex edge features z = pe[src] * conj(pe[dst]).
// One wave per edge (32 lanes = 32 (q,m) pairs); pe loads fully coalesced.
// ---------------------------------------------------------------------------
__global__ void zfeat_kernel(const float* __restrict__ pe, const int* __restrict__ src,
                             const int* __restrict__ dst, const int* __restrict__ batch,
                             float* __restrict__ z_re, float* __restrict__ z_im,
                             int* __restrict__ eg) {
  int tid = blockIdx.x * 256 + threadIdx.x;   // E_*32 threads total
  int e = tid >> 5;
  int j = tid & 31;
  if (e >= E_) return;
  int s = src[e], d = dst[e];
  v2f zs = *(const v2f*)&pe[(size_t)s * 64 + j * 2];
  v2f zd = *(const v2f*)&pe[(size_t)d * 64 + j * 2];
  z_re[(size_t)e * 32 + j] = zs.x * zd.x + zs.y * zd.y;
  z_im[(size_t)e * 32 + j] = zs.y * zd.x - zs.x * zd.y;
  if (j == 0) eg[e] = batch[s];
}

// ---------------------------------------------------------------------------
// Tiny eigenvalue MLP: scalar -> 16 -> scalar, per (B,Q,M) entry.
// ---------------------------------------------------------------------------
__global__ void phi_kernel(const float* __restrict__ Lam, const float* __restrict__ w1,
                           const float* __restrict__ b1, const float* __restrict__ w2,
                           const float* __restrict__ b2, float* __restrict__ wout) {
  int i = blockIdx.x * 256 + threadIdx.x;
  if (i >= B_ * 32) return;
  float lam = Lam[i];
  float acc = b2[0];
#pragma unroll
  for (int j = 0; j < 16; ++j) {
    float h = fmaf(lam, w1[j], b1[j]);
    h = fmaxf(h, 0.0f);
    acc = fmaf(h, w2[j], acc);
  }
  wout[i] = acc;
}

__global__ void zero_kernel(float* __restrict__ p, int n) {
  int i = blockIdx.x * 256 + threadIdx.x;
  if (i < n) p[i] = 0.0f;
}

__global__ void copy_kernel(const float* __restrict__ s, float* __restrict__ d, int n) {
  int i = blockIdx.x * 256 + threadIdx.x;
  if (i < n) d[i] = s[i];
}

// ---------------------------------------------------------------------------
// Edge kernel: per 16-edge tile, fused
//   feat = [z_re*we, z_im*we]           (LDS, 16x64)
//   edge_pe = feat @ enc_w + enc_b      (WMMA f32 16x16x4, K=64)
//   ea = edge_pe + edge_attr            (LDS, 16x64)
//   msg = relu(x[src] + ea @ lin_w + lin_b)
//   agg[dst] += msg                     (global f32 atomics)
// Block = 128 threads = 4 waves; wave w owns output columns [16w,16w+16).
// Weight B-fragments preloaded once per block; grid-stride over tiles.
// ---------------------------------------------------------------------------
__global__ void __launch_bounds__(128)
edge_kernel(const float* __restrict__ z_re, const float* __restrict__ z_im,
            const int* __restrict__ eg, const float* __restrict__ wq,
            const float* __restrict__ encw, const float* __restrict__ encb,
            const float* __restrict__ linw, const float* __restrict__ linb,
            const float* __restrict__ edge_attr, const float* __restrict__ xcur,
            const int* __restrict__ src, const int* __restrict__ dst,
            float* __restrict__ agg, int ntiles) {
  __shared__ float featA[16 * 66];
  __shared__ float eaB[16 * 66];
  __shared__ int sT[16], dT[16], egT[16];

  const int t = threadIdx.x;
  const int lane = t & 31;
  const int wv = t >> 5;     // wave in block: 0..3
  const int g = lane >> 4;   // half-wave
  const int nn = lane & 15;
  const int col = wv * 16 + nn;

  // Preload B-fragments for both 64x64 weight matrices (ISA B layout:
  // lane g*16+n holds {B[2g][n], B[2g+1][n]} per K-step of 4).
  v2f B1[16], B2[16];
#pragma unroll
  for (int k = 0; k < 16; ++k) {
    int r = 4 * k + 2 * g;
    v2f b;
    b.x = encw[r * 64 + col];
    b.y = encw[(r + 1) * 64 + col];
    B1[k] = b;
    b.x = linw[r * 64 + col];
    b.y = linw[(r + 1) * 64 + col];
    B2[k] = b;
  }
  const float encb_c = encb[col];
  const float linb_c = linb[col];

  for (int tile = blockIdx.x; tile < ntiles; tile += gridDim.x) {
    const int e0 = tile * 16;
    if (t < 16)      sT[t] = src[e0 + t];
    else if (t < 32) dT[t - 16] = dst[e0 + t - 16];
    else if (t < 48) egT[t - 32] = eg[e0 + t - 32];
    __syncthreads();

    // Build feat tile [16 x 64] in LDS.
#pragma unroll
    for (int i = 0; i < 8; ++i) {
      int idx = t + i * 128;
      int m = idx >> 6;
      int j = idx & 63;
      int jj = j & 31;
      int e = e0 + m;
      float wval = wq[egT[m] * 32 + jj];
      float zval = (j < 32) ? z_re[(size_t)e * 32 + jj] : z_im[(size_t)e * 32 + jj];
      featA[m * 66 + j] = zval * wval;
    }
    __syncthreads();

    // GEMM1: feat @ enc_w (this wave's 16 columns), K=64 -> 16 WMMAs.
    v8f acc = {};
#pragma unroll
    for (int k = 0; k < 16; ++k) {
      v2f a = *(const v2f*)&featA[nn * 66 + 4 * k + 2 * g];
      acc = WMMA_F32X4(a, B1[k], acc);
    }
    // ea = edge_pe + enc_b + edge_attr -> LDS (D layout: VGPR r = row 8g+r).
#pragma unroll
    for (int r = 0; r < 8; ++r) {
      int m = 8 * g + r;
      float v = acc[r] + encb_c + edge_attr[(size_t)(e0 + m) * 64 + col];
      eaB[m * 66 + col] = v;
    }
    __syncthreads();

    // GEMM2: ea @ lin_w.
    v8f acc2 = {};
#pragma unroll
    for (int k = 0; k < 16; ++k) {
      v2f a = *(const v2f*)&eaB[nn * 66 + 4 * k + 2 * g];
      acc2 = WMMA_F32X4(a, B2[k], acc2);
    }
    // msg = relu(x[src] + .); scatter-add into agg[dst].
#pragma unroll
    for (int r = 0; r < 8; ++r) {
      int m = 8 * g + r;
      float v = acc2[r] + linb_c + xcur[(size_t)sT[m] * 64 + col];
      v = fmaxf(v, 0.0f);
      atomicAdd(&agg[(size_t)dT[m] * 64 + col], v);
    }
    __syncthreads();
  }
}

// ---------------------------------------------------------------------------
// Node kernel: per 16-row tile, in -> in @ W + bias, store, and accumulate
// per-column sum / sum-of-squares for the following BatchNorm.
// mode 0: in = xcur + agg
// mode 1: in = relu(bn1(t1)) using (mu, rs, gam, bet)
// ---------------------------------------------------------------------------
__global__ void __launch_bounds__(128)
node_kernel(const float* __restrict__ inA, const float* __restrict__ inB,
            const float* __restrict__ mu, const float* __restrict__ rs,
            const float* __restrict__ gam, const float* __restrict__ bet,
            const float* __restrict__ W, const float* __restrict__ bias,
            float* __restrict__ outT, float* __restrict__ gsum,
            float* __restrict__ gsq, int ntiles, int mode) {
  __shared__ float hT[16 * 66];
  __shared__ float smSum[64], smSq[64];

  const int t = threadIdx.x;
  const int lane = t & 31;
  const int wv = t >> 5;
  const int g = lane >> 4;
  const int nn = lane & 15;
  const int col = wv * 16 + nn;

  if (t < 64) smSum[t] = 0.0f;
  else smSq[t - 64] = 0.0f;

  v2f Bf[16];
#pragma unroll
  for (int k = 0; k < 16; ++k) {
    int r = 4 * k + 2 * g;
    v2f b;
    b.x = W[r * 64 + col];
    b.y = W[(r + 1) * 64 + col];
    Bf[k] = b;
  }
  const float bias_c = bias[col];
  float ls = 0.0f, lsq = 0.0f;

  for (int tile = blockIdx.x; tile < ntiles; tile += gridDim.x) {
    const int r0 = tile * 16;
    __syncthreads();   // guard hT WAR vs previous iteration's WMMA reads
#pragma unroll
    for (int i = 0; i < 8; ++i) {
      int idx = t + i * 128;
      int m = idx >> 6;
      int j = idx & 63;
      size_t gi = (size_t)(r0 + m) * 64 + j;
      float v;
      if (mode == 0) {
        v = inA[gi] + inB[gi];
      } else {
        v = (inA[gi] - mu[j]) * rs[j] * gam[j] + bet[j];
        v = fmaxf(v, 0.0f);
      }
      hT[m * 66 + j] = v;
    }
    __syncthreads();

    v8f acc = {};
#pragma unroll
    for (int k = 0; k < 16; ++k) {
      v2f a = *(const v2f*)&hT[nn * 66 + 4 * k + 2 * g];
      acc = WMMA_F32X4(a, Bf[k], acc);
    }
#pragma unroll
    for (int r = 0; r < 8; ++r) {
      int m = 8 * g + r;
      float v = acc[r] + bias_c;
      outT[(size_t)(r0 + m) * 64 + col] = v;
      ls += v;
      lsq += v * v;
    }
  }
  __syncthreads();
  atomicAdd(&smSum[col], ls);
  atomicAdd(&smSq[col], lsq);
  __syncthreads();
  if (t < 64) atomicAdd(&gsum[t], smSum[t]);
  else atomicAdd(&gsq[t - 64], smSq[t - 64]);
}

__global__ void bnstat_kernel(const float* __restrict__ sum, const float* __restrict__ sq,
                              float* __restrict__ mu, float* __restrict__ rs, float invN) {
  int j = threadIdx.x;  // 64 threads
  float m = sum[j] * invN;
  float v = sq[j] * invN - m * m;
  mu[j] = m;
  rs[j] = rsqrtf(v + 1e-5f);
}

__global__ void bnapply_kernel(const float* __restrict__ tin, const float* __restrict__ mu,
                               const float* __restrict__ rs, const float* __restrict__ gam,
                               const float* __restrict__ bet, float* __restrict__ out,
                               int n, int dorelu) {
  int i = blockIdx.x * 256 + threadIdx.x;
  if (i >= n) return;
  int j = i & 63;
  float v = (tin[i] - mu[j]) * rs[j] * gam[j] + bet[j];
  if (dorelu) v = fmaxf(v, 0.0f);
  out[i] = v;
}

// ---------------------------------------------------------------------------

extern "C" void kernel_launch(void* const* d_in, const int* in_sizes, int n_in,
                              void* d_out, int out_size, void* d_ws, size_t ws_size,
                              hipStream_t stream) {
  const float* x      = (const float*)d_in[0];
  const float* pe     = (const float*)d_in[1];
  const float* Lam    = (const float*)d_in[2];
  const float* eattr  = (const float*)d_in[3];
  const float* phi_w1 = (const float*)d_in[4];
  const float* phi_b1 = (const float*)d_in[5];
  const float* phi_w2 = (const float*)d_in[6];
  const float* phi_b2 = (const float*)d_in[7];
  const float* enc_w  = (const float*)d_in[8];
  const float* enc_b  = (const float*)d_in[9];
  const float* lin_w  = (const float*)d_in[10];
  const float* lin_b  = (const float*)d_in[11];
  const float* mlp_w1 = (const float*)d_in[12];
  const float* mlp_b1 = (const float*)d_in[13];
  const float* bn1_g  = (const float*)d_in[14];
  const float* bn1_b  = (const float*)d_in[15];
  const float* mlp_w2 = (const float*)d_in[16];
  const float* mlp_b2 = (const float*)d_in[17];
  const float* bn2_g  = (const float*)d_in[18];
  const float* bn2_b  = (const float*)d_in[19];
  const int* eidx     = (const int*)d_in[20];
  const int* batch    = (const int*)d_in[21];
  float* out = (float*)d_out;

  const int* src = eidx;
  const int* dst = eidx + E_;

  // Workspace layout (floats).
  float* ws = (float*)d_ws;
  size_t o = 0;
  float* z_re = ws + o;  o += (size_t)E_ * 32;
  float* z_im = ws + o;  o += (size_t)E_ * 32;
  int*   eg   = (int*)(ws + o);  o += E_;
  float* wq   = ws + o;  o += 4096;
  float* xcur = ws + o;  o += (size_t)N_ * 64;
  float* agg  = ws + o;  o += (size_t)N_ * 64;   // agg + stats zeroed together
  float* stats= ws + o;  o += 256;               // sum1, sq1, sum2, sq2
  float* musig= ws + o;  o += 256;               // mu1, rs1, mu2, rs2
  float* t1   = ws + o;  o += (size_t)N_ * 64;
  float* t2   = ws + o;  o += (size_t)N_ * 64;

  const int NH = N_ * 64;                        // 6,400,000
  const int etiles = E_ / 16;                    // 50,000 (exact)
  const int ntiles = N_ / 16;                    // 6,250  (exact)

  // One-time: invariant edge features + x working copy + pe output copy.
  zfeat_kernel<<<(E_ * 32) / 256, 256, 0, stream>>>(pe, src, dst, batch, z_re, z_im, eg);
  copy_kernel<<<NH / 256, 256, 0, stream>>>(x, xcur, NH);
  copy_kernel<<<NH / 256, 256, 0, stream>>>(pe, out + (size_t)NH, NH);

  for (int l = 0; l < L_; ++l) {
    phi_kernel<<<(B_ * 32 + 255) / 256, 256, 0, stream>>>(
        Lam, phi_w1 + l * 16, phi_b1 + l * 16, phi_w2 + l * 16, phi_b2 + l, wq);
    zero_kernel<<<(NH + 256 + 255) / 256, 256, 0, stream>>>(agg, NH + 256);

    edge_kernel<<<2048, 128, 0, stream>>>(
        z_re, z_im, eg, wq,
        enc_w + (size_t)l * 64 * 64, enc_b + l * 64,
        lin_w + (size_t)l * 64 * 64, lin_b + l * 64,
        eattr, xcur, src, dst, agg, etiles);

    // h = x + agg; t1 = h @ mlp_w1 + b1; bn1 stats.
    node_kernel<<<625, 128, 0, stream>>>(
        xcur, agg, musig, musig + 64, bn1_g + l * 64, bn1_b + l * 64,
        mlp_w1 + (size_t)l * 64 * 64, mlp_b1 + l * 64,
        t1, stats, stats + 64, ntiles, 0);
    bnstat_kernel<<<1, 64, 0, stream>>>(stats, stats + 64, musig, musig + 64,
                                        1.0f / (float)N_);

    // u = relu(bn1(t1)); t2 = u @ mlp_w2 + b2; bn2 stats.
    node_kernel<<<625, 128, 0, stream>>>(
        t1, t1, musig, musig + 64, bn1_g + l * 64, bn1_b + l * 64,
        mlp_w2 + (size_t)l * 64 * 64, mlp_b2 + l * 64,
        t2, stats + 128, stats + 192, ntiles, 1);
    bnstat_kernel<<<1, 64, 0, stream>>>(stats + 128, stats + 192,
                                        musig + 128, musig + 192, 1.0f / (float)N_);

    // x = bn2(t2) (+relu except last layer); last layer writes d_out.
    float* target = (l == L_ - 1) ? out : xcur;
    bnapply_kernel<<<NH / 256, 256, 0, stream>>>(
        t2, musig + 128, musig + 192, bn2_g + l * 64, bn2_b + l * 64,
        target, NH, (l != L_ - 1) ? 1 : 0);
  }
}